// AdditiveAttention_79534204387517
// MI455X (gfx1250) — compile-verified
//
#include <hip/hip_runtime.h>
#include <hip/hip_bf16.h>

// Problem constants (from reference): B=32, S=2048, H=1024, fp32 throughout.
#define BATCH 32
#define SEQ   2048
#define HID   1024
#define TILE_S 128          // rows of lstm per block in the scores kernel
#define BK     32           // K-chunk staged in LDS per iteration
#define LDS_STRIDE 36       // 32 cols + 4 pad floats (keeps 16B alignment, kills bank conflicts)

typedef __attribute__((ext_vector_type(2))) float v2f;
typedef __attribute__((ext_vector_type(8))) float v8f;

// ---------------------------------------------------------------------------
// Kernel 1: scores[b,s] = lstm[b,s,:] . w1   via V_WMMA_F32_16X16X4_F32.
// Block = 256 threads (8 wave32s). blockIdx = (s-tile, batch).
// Each wave owns 16 consecutive rows; w1 is replicated across the 16 B-matrix
// columns, so every column of the 16x16 f32 accumulator holds the scores.
// ---------------------------------------------------------------------------
__global__ __launch_bounds__(256)
void aa_scores_wmma(const float* __restrict__ lstm,
                    const float* __restrict__ W,   // [2*H], w1 = W[0:H]
                    float* __restrict__ scores)    // [B*S] (attn slot of d_out)
{
    __shared__ float ldsA[TILE_S * LDS_STRIDE];   // 128 x 32 fp32 tile (padded)
    __shared__ float ldsW[HID];                   // w1 staged once

    const int tid  = threadIdx.x;
    const int wave = tid >> 5;
    const int lane = tid & 31;
    const int half = lane >> 4;        // K-group select (ISA A-matrix layout)
    const int m    = lane & 15;        // row within the wave's 16-row strip
    const int b    = blockIdx.y;
    const int s0   = blockIdx.x * TILE_S;

    // stage w1 into LDS
    for (int i = tid; i < HID; i += 256) ldsW[i] = W[i];

    const int   arow = wave * 16 + m;                       // LDS row this lane reads
    const int   ldrow = tid >> 1;                           // cooperative-load row
    const int   part  = tid & 1;                            // 16-float half of the chunk
    const float* gsrc_base = lstm + ((size_t)(b * SEQ + s0 + ldrow)) * HID + part * 16;

    v8f acc = {};   // 16x16 f32 accumulator (8 VGPRs)

    for (int kk = 0; kk < HID; kk += BK) {
        __syncthreads();   // protect ldsA from previous iteration's readers
        // cooperative load: 128 rows x 32 cols, 2 threads/row, 4x float4 each
        {
            const float4* g = (const float4*)(gsrc_base + kk);
            float4* l = (float4*)&ldsA[ldrow * LDS_STRIDE + part * 16];
            #pragma unroll
            for (int j = 0; j < 4; ++j) l[j] = g[j];
            if (kk + BK < HID)
                __builtin_prefetch(gsrc_base + kk + BK, 0, 1);  // global_prefetch_b8
        }
        __syncthreads();

        // 8 WMMAs cover K = kk .. kk+31
        #pragma unroll
        for (int k = 0; k < BK / 4; ++k) {
            const int kc = k * 4 + 2 * half;   // this lane's K pair (ISA layout)
            v2f a, w;
            a.x = ldsA[arow * LDS_STRIDE + kc];
            a.y = ldsA[arow * LDS_STRIDE + kc + 1];
            w.x = ldsW[kk + kc];
            w.y = ldsW[kk + kc + 1];
            acc = __builtin_amdgcn_wmma_f32_16x16x4_f32(
                      /*neg_a=*/false, a, /*neg_b=*/false, w,
                      /*c_mod=*/(short)0, acc,
                      /*reuse_a=*/false, /*reuse_b=*/false);
        }
    }

    // Column 0 (lanes 0 and 16 of each wave) carries all 16 row scores:
    // VGPR j: lanes 0-15 -> M=j, lanes 16-31 -> M=8+j.
    if (m == 0) {
        const int mb = wave * 16 + half * 8;
        float* out = scores + (size_t)b * SEQ + s0 + mb;
        #pragma unroll
        for (int j = 0; j < 8; ++j) out[j] = acc[j];
    }
}

// ---------------------------------------------------------------------------
// Kernel 2: per-batch softmax (in place).  Also adds hb = fh.w2 + bias first
// (softmax-invariant, but matches the reference arithmetic).
// One block per batch, 256 threads, 8 elements/thread.
// ---------------------------------------------------------------------------
__global__ __launch_bounds__(256)
void aa_softmax(float* __restrict__ attn,           // [B*S], holds raw scores on entry
                const float* __restrict__ fh,       // [B*H]
                const float* __restrict__ W,        // [2*H], w2 = W[H:2H]
                const float* __restrict__ bias)     // [1]
{
    __shared__ float red[256];
    const int tid = threadIdx.x;
    const int b   = blockIdx.x;

    // hb = dot(final_hidden[b], w2) + bias
    float p = 0.f;
    for (int i = tid; i < HID; i += 256) p += fh[(size_t)b * HID + i] * W[HID + i];
    red[tid] = p; __syncthreads();
    for (int s = 128; s > 0; s >>= 1) { if (tid < s) red[tid] += red[tid + s]; __syncthreads(); }
    const float hb = red[0] + bias[0];
    __syncthreads();

    float x[SEQ / 256];
    float mx = -3.0e38f;
    float* row = attn + (size_t)b * SEQ;
    #pragma unroll
    for (int j = 0; j < SEQ / 256; ++j) {
        x[j] = row[tid + j * 256] + hb;
        mx = fmaxf(mx, x[j]);
    }
    red[tid] = mx; __syncthreads();
    for (int s = 128; s > 0; s >>= 1) { if (tid < s) red[tid] = fmaxf(red[tid], red[tid + s]); __syncthreads(); }
    mx = red[0]; __syncthreads();

    float sum = 0.f;
    #pragma unroll
    for (int j = 0; j < SEQ / 256; ++j) { x[j] = expf(x[j] - mx); sum += x[j]; }
    red[tid] = sum; __syncthreads();
    for (int s = 128; s > 0; s >>= 1) { if (tid < s) red[tid] += red[tid + s]; __syncthreads(); }
    const float inv = 1.0f / red[0];

    #pragma unroll
    for (int j = 0; j < SEQ / 256; ++j) row[tid + j * 256] = x[j] * inv;
}

// ---------------------------------------------------------------------------
// Kernel 3a: zero the context output region (harness poisons d_out).
// ---------------------------------------------------------------------------
__global__ __launch_bounds__(256)
void aa_zero_ctx(float4* __restrict__ ctx4)   // BATCH*HID/4 float4s
{
    ctx4[blockIdx.x * 256 + threadIdx.x] = float4{0.f, 0.f, 0.f, 0.f};
}

// ---------------------------------------------------------------------------
// Kernel 3b: context[b,h] = sum_s attn[b,s] * lstm[b,s,h].
// Grid (B, SEQ/256): each block streams a 256-row slice of one batch
// (fully coalesced float4 loads) and accumulates via global fp32 atomics
// -> 256 blocks keep HBM saturated.
// ---------------------------------------------------------------------------
__global__ __launch_bounds__(256)
void aa_context(const float* __restrict__ lstm,
                const float* __restrict__ attn,   // [B*S]
                float* __restrict__ ctx)          // [B*H]
{
    const int tid = threadIdx.x;
    const int b   = blockIdx.x;
    const int s0  = blockIdx.y * 256;

    const float*  arow = attn + (size_t)b * SEQ + s0;
    const float4* l4   = (const float4*)(lstm + ((size_t)(b * SEQ + s0)) * HID);

    float4 acc = {0.f, 0.f, 0.f, 0.f};
    #pragma unroll 4
    for (int s = 0; s < 256; ++s) {
        const float  w = arow[s];                   // uniform -> scalar load
        const float4 v = l4[(size_t)s * (HID / 4) + tid];
        acc.x += w * v.x;
        acc.y += w * v.y;
        acc.z += w * v.z;
        acc.w += w * v.w;
    }

    float* c = ctx + (size_t)b * HID + tid * 4;
    unsafeAtomicAdd(c + 0, acc.x);   // global_atomic_add_f32
    unsafeAtomicAdd(c + 1, acc.y);
    unsafeAtomicAdd(c + 2, acc.z);
    unsafeAtomicAdd(c + 3, acc.w);
}

// ---------------------------------------------------------------------------
extern "C" void kernel_launch(void* const* d_in, const int* in_sizes, int n_in,
                              void* d_out, int out_size, void* d_ws, size_t ws_size,
                              hipStream_t stream)
{
    (void)in_sizes; (void)n_in; (void)out_size; (void)d_ws; (void)ws_size;

    const float* lstm = (const float*)d_in[0];   // [B,S,H]
    const float* fh   = (const float*)d_in[1];   // [B,H]
    const float* W    = (const float*)d_in[2];   // [1,2H]
    const float* bias = (const float*)d_in[3];   // [1]

    float* ctx  = (float*)d_out;                 // [B,H]   first output
    float* attn = (float*)d_out + BATCH * HID;   // [B,S]   second output (scores in-place)

    // 1) zero context accumulator region
    aa_zero_ctx<<<dim3((BATCH * HID / 4) / 256), dim3(256), 0, stream>>>((float4*)ctx);

    // 2) scores via f32 WMMA -> attn slot
    aa_scores_wmma<<<dim3(SEQ / TILE_S, BATCH), dim3(256), 0, stream>>>(lstm, W, attn);

    // 3) softmax in place (adds fh.w2 + bias first)
    aa_softmax<<<dim3(BATCH), dim3(256), 0, stream>>>(attn, fh, W, bias);

    // 4) context accumulation
    aa_context<<<dim3(BATCH, SEQ / 256), dim3(256), 0, stream>>>(lstm, attn, ctx);
}